// GCN_Net_82824149336812
// MI455X (gfx1250) — compile-verified
//
#include <hip/hip_runtime.h>
#include <hip/hip_bf16.h>

#define N_NODES 200000
#define N_EDGES 6400000
#define EMB 32
#define HID 16
#define BY 4
#define NSYMP 15
#define NOPT 3
#define XCOLS (BY + 1 + NSYMP)   // 20 columns in x

typedef __attribute__((ext_vector_type(2))) float v2f;
typedef __attribute__((ext_vector_type(8))) float v8f;

// ---------------------------------------------------------------------------
// Phase 1: embedding assembly. One lane per (node, dim): 32 lanes per node,
// table gathers are coalesced across lanes (tables are ~6.5KB, cache-resident).
// ---------------------------------------------------------------------------
__global__ void embed_kernel(const int* __restrict__ x,
                             const float* __restrict__ birth_table,
                             const float* __restrict__ gender_table,
                             const float* __restrict__ symp_tables,
                             float* __restrict__ h0) {
    int t = blockIdx.x * blockDim.x + threadIdx.x;
    int node = t >> 5;
    int d    = t & 31;
    if (node >= N_NODES) return;
    const int* xr = x + (size_t)node * XCOLS;
    // argmax over the one-hot birth columns (first max wins, like jnp.argmax)
    int bi = 0, best = xr[0];
#pragma unroll
    for (int j = 1; j < BY; ++j) { int v = xr[j]; if (v > best) { best = v; bi = j; } }
    int g = xr[BY];
    float acc = birth_table[bi * EMB + d] + gender_table[g * EMB + d];
    float s = 0.0f;
#pragma unroll
    for (int j = 0; j < NSYMP; ++j) {
        int opt = xr[BY + 1 + j];
        s += symp_tables[((size_t)j * NOPT + opt) * EMB + d];
    }
    acc += s * (1.0f / (float)NSYMP);
    h0[(size_t)node * EMB + d] = acc * (1.0f / 3.0f);
}

// ---------------------------------------------------------------------------
// Degree: deg starts at 1 (self loop), atomic count of incoming edges,
// then converted to rsqrt in place.
// ---------------------------------------------------------------------------
__global__ void init_deg_kernel(float* __restrict__ deg) {
    int i = blockIdx.x * blockDim.x + threadIdx.x;
    if (i < N_NODES) deg[i] = 1.0f;
}

__global__ void deg_count_kernel(const int* __restrict__ tgt, float* __restrict__ deg) {
    int e = blockIdx.x * blockDim.x + threadIdx.x;
    if (e < N_EDGES) atomicAdd(&deg[tgt[e]], 1.0f);
}

__global__ void deg_rsqrt_kernel(float* __restrict__ deg) {
    int i = blockIdx.x * blockDim.x + threadIdx.x;
    if (i < N_NODES) deg[i] = rsqrtf(deg[i]);   // deg >= 1 always (self loop)
}

// ---------------------------------------------------------------------------
// Dense GEMM via V_WMMA_F32_16X16X4_F32. One wave32 computes a 16-node x 16-col
// output tile, looping K in chunks of 4.
// A (16x4 fp32) VGPR layout: lanes 0-15 hold K=k0,k0+1 ; lanes 16-31 K=k0+2,k0+3.
// B (4x16 fp32): VGPR0 = row K (lanes 0-15) / K+2 (lanes 16-31), VGPR1 = K+1/K+3.
// D (16x16 fp32): VGPR r -> row r (lanes 0-15) / row r+8 (lanes 16-31).
// ---------------------------------------------------------------------------
template <int K>
__global__ void gemm_wmma_kernel(const float* __restrict__ A,   // [N_NODES, K]
                                 const float* __restrict__ W,   // [K, 16]
                                 float* __restrict__ out) {     // [N_NODES, 16]
    int wave = threadIdx.x >> 5;
    int lane = threadIdx.x & 31;
    int tile = blockIdx.x * (blockDim.x >> 5) + wave;
    if (tile * 16 >= N_NODES) return;          // wave-uniform: EXEC stays all-1s
    int m0   = tile * 16;
    int half = lane >> 4;                      // 0: lanes 0-15, 1: lanes 16-31
    int l    = lane & 15;

    v8f c = {0.f, 0.f, 0.f, 0.f, 0.f, 0.f, 0.f, 0.f};
#pragma unroll
    for (int k0 = 0; k0 < K; k0 += 4) {
        int kk = k0 + 2 * half;
        const float* ap = A + (size_t)(m0 + l) * K + kk;
        v2f a, b;
        a.x = ap[0];
        a.y = ap[1];
        b.x = W[kk * HID + l];
        b.y = W[(kk + 1) * HID + l];
        c = __builtin_amdgcn_wmma_f32_16x16x4_f32(
                /*neg_a=*/false, a, /*neg_b=*/false, b,
                /*c_mod=*/(short)0, c, /*reuse_a=*/false, /*reuse_b=*/false);
    }

    float* op = out + (size_t)(m0 + 8 * half) * HID + l;
#pragma unroll
    for (int r = 0; r < 8; ++r) op[(size_t)r * HID] = c[r];
}

// ---------------------------------------------------------------------------
// Zero an fp32 buffer.
// ---------------------------------------------------------------------------
__global__ void zero_kernel(float* __restrict__ p, int n) {
    int i = blockIdx.x * blockDim.x + threadIdx.x;
    if (i < n) p[i] = 0.0f;
}

// ---------------------------------------------------------------------------
// Edge scatter: agg[tgt] += dinv[src]*dinv[tgt] * msg[src].  float4 gathers,
// 16 global f32 atomics per edge.
// ---------------------------------------------------------------------------
__global__ void scatter_kernel(const int* __restrict__ src,
                               const int* __restrict__ tgt,
                               const float* __restrict__ dinv,
                               const float* __restrict__ msg,   // [N,16]
                               float* __restrict__ agg) {       // [N,16]
    int e = blockIdx.x * blockDim.x + threadIdx.x;
    if (e >= N_EDGES) return;
    int s = src[e];
    int t = tgt[e];
    float norm = dinv[s] * dinv[t];
    const float4* mp = (const float4*)(msg + (size_t)s * HID);
    float* ap = agg + (size_t)t * HID;
#pragma unroll
    for (int q = 0; q < 4; ++q) {
        float4 v = mp[q];
        atomicAdd(ap + 4 * q + 0, norm * v.x);
        atomicAdd(ap + 4 * q + 1, norm * v.y);
        atomicAdd(ap + 4 * q + 2, norm * v.z);
        atomicAdd(ap + 4 * q + 3, norm * v.w);
    }
}

// ---------------------------------------------------------------------------
// Layer-1 finish (in place over agg): h1 = ELU(agg + dinv^2 * t1 + b1)
// ---------------------------------------------------------------------------
__global__ void finish1_kernel(const float* __restrict__ t1,
                               const float* __restrict__ dinv,
                               const float* __restrict__ b1,
                               float* __restrict__ agg) {
    int i = blockIdx.x * blockDim.x + threadIdx.x;
    if (i >= N_NODES * HID) return;
    int node = i >> 4;
    int f    = i & 15;
    float di = dinv[node];
    float v  = agg[i] + di * di * t1[i] + b1[f];
    agg[i] = (v > 0.0f) ? v : (expf(v) - 1.0f);   // ELU, alpha=1
}

// ---------------------------------------------------------------------------
// Layer-2 finish fused with final projection:
// out[n] = sum_f (agg2[n,f] + dinv[n]^2*t2[n,f] + b2[f]) * Wl[f] + bl
// ---------------------------------------------------------------------------
__global__ void out_kernel(const float* __restrict__ agg2,
                           const float* __restrict__ t2,
                           const float* __restrict__ dinv,
                           const float* __restrict__ b2,
                           const float* __restrict__ Wl,
                           const float* __restrict__ bl,
                           float* __restrict__ out) {
    int node = blockIdx.x * blockDim.x + threadIdx.x;
    if (node >= N_NODES) return;
    float di  = dinv[node];
    float di2 = di * di;
    float acc = bl[0];
    const float* ar = agg2 + (size_t)node * HID;
    const float* tr = t2   + (size_t)node * HID;
#pragma unroll
    for (int f = 0; f < HID; ++f) {
        float v = ar[f] + di2 * tr[f] + b2[f];
        acc += v * Wl[f];
    }
    out[node] = acc;
}

// ---------------------------------------------------------------------------
// Host-side orchestration.
// Inputs (setup_inputs order):
//  0: x [N,20] int32          1: edge_index [2,E] int32
//  2: birth_table [4,32] f32  3: gender_table [2,32] f32
//  4: symp_tables [15,3,32]   5: W1 [32,16]   6: b1 [16]
//  7: W2 [16,16]  8: b2 [16]  9: Wl [16,1]   10: bl [1]
// Output: [N,1] f32
// Workspace layout (floats):
//  dinv : N            h0 : N*32 (reused as agg2 in layer 2)
//  t1   : N*16 (reused as t2)    agg1 : N*16 (becomes h1 in place)
// Total 52 MB.
// ---------------------------------------------------------------------------
extern "C" void kernel_launch(void* const* d_in, const int* in_sizes, int n_in,
                              void* d_out, int out_size, void* d_ws, size_t ws_size,
                              hipStream_t stream) {
    const int*   x            = (const int*)d_in[0];
    const int*   edge_index   = (const int*)d_in[1];
    const float* birth_table  = (const float*)d_in[2];
    const float* gender_table = (const float*)d_in[3];
    const float* symp_tables  = (const float*)d_in[4];
    const float* W1 = (const float*)d_in[5];
    const float* b1 = (const float*)d_in[6];
    const float* W2 = (const float*)d_in[7];
    const float* b2 = (const float*)d_in[8];
    const float* Wl = (const float*)d_in[9];
    const float* bl = (const float*)d_in[10];
    float* out = (float*)d_out;

    const int* src = edge_index;              // row 0
    const int* tgt = edge_index + N_EDGES;    // row 1

    float* ws   = (float*)d_ws;
    float* dinv = ws;
    float* h0   = dinv + N_NODES;             // N*32 (doubles as agg2 later)
    float* t1   = h0 + (size_t)N_NODES * EMB; // N*16 (doubles as t2)
    float* agg1 = t1 + (size_t)N_NODES * HID; // N*16 (becomes h1 in place)

    const int TB = 256;
    const int node_blocks    = (N_NODES + TB - 1) / TB;                 // 782
    const int nodedim_blocks = (N_NODES * EMB + TB - 1) / TB;           // 25000
    const int row16_blocks   = (N_NODES * HID + TB - 1) / TB;           // 12500
    const int edge_blocks    = (N_EDGES + TB - 1) / TB;                 // 25000
    const int tiles          = N_NODES / 16;                            // 12500
    const int gemm_blocks    = (tiles + (TB / 32) - 1) / (TB / 32);     // 1563

    // Phase 1: embeddings
    embed_kernel<<<nodedim_blocks, TB, 0, stream>>>(x, birth_table, gender_table,
                                                    symp_tables, h0);
    // Degree -> dinv
    init_deg_kernel<<<node_blocks, TB, 0, stream>>>(dinv);
    deg_count_kernel<<<edge_blocks, TB, 0, stream>>>(tgt, dinv);
    deg_rsqrt_kernel<<<node_blocks, TB, 0, stream>>>(dinv);

    // Layer 1
    gemm_wmma_kernel<EMB><<<gemm_blocks, TB, 0, stream>>>(h0, W1, t1);
    zero_kernel<<<row16_blocks, TB, 0, stream>>>(agg1, N_NODES * HID);
    scatter_kernel<<<edge_blocks, TB, 0, stream>>>(src, tgt, dinv, t1, agg1);
    finish1_kernel<<<row16_blocks, TB, 0, stream>>>(t1, dinv, b1, agg1); // agg1 -> h1

    // Layer 2 (t2 reuses t1 buffer, agg2 reuses h0 buffer)
    float* h1   = agg1;
    float* t2   = t1;
    float* agg2 = h0;
    gemm_wmma_kernel<HID><<<gemm_blocks, TB, 0, stream>>>(h1, W2, t2);
    zero_kernel<<<row16_blocks, TB, 0, stream>>>(agg2, N_NODES * HID);
    scatter_kernel<<<edge_blocks, TB, 0, stream>>>(src, tgt, dinv, t2, agg2);

    // Final projection
    out_kernel<<<node_blocks, TB, 0, stream>>>(agg2, t2, dinv, b2, Wl, bl, out);
}